// GNN_87806311399724
// MI455X (gfx1250) — compile-verified
//
#include <hip/hip_runtime.h>
#include <math.h>
#include <stdint.h>

// ---------------------------------------------------------------------------
// GINEConv stack for MI455X (gfx1250, wave32).
// Roofline: edge phase moves ~4.3 GB total (gather x[src], edge_attr, scatter
// atomics) -> ~185us @ 23.3 TB/s; GEMMs are only 28 GFLOP. So the edge phase
// owns HBM/VMEM, and the fused MLP kernel keeps its inner loop entirely on
// LDS + V_WMMA_F32_16X16X4_F32: weights are staged into LDS by the Tensor
// Data Mover (TENSOR_LOAD_TO_LDS, TENSORcnt), overlapped with the cooperative
// (x+agg) tile load.
// ---------------------------------------------------------------------------

#define D      128
#define MB     64          // rows (nodes) per MLP block
#define LSTR   132         // LDS row stride (D+4): bank-shift 4/row, 528B keeps alignment
#define LN_EPS 1e-5f

typedef float v2f __attribute__((ext_vector_type(2)));
typedef float v8f __attribute__((ext_vector_type(8)));
typedef unsigned int v4u __attribute__((ext_vector_type(4)));
typedef int v4i __attribute__((ext_vector_type(4)));
typedef int v8i __attribute__((ext_vector_type(8)));

__device__ __forceinline__ float gelu_erf(float v) {
    return 0.5f * v * (1.0f + erff(v * 0.70710678118654752440f));
}

#if __has_builtin(__builtin_amdgcn_s_wait_tensorcnt)
#define WAIT_TENSORCNT0() __builtin_amdgcn_s_wait_tensorcnt(0)
#else
#define WAIT_TENSORCNT0() asm volatile("s_wait_tensorcnt 0x0" ::: "memory")
#endif

#if __has_builtin(__builtin_amdgcn_tensor_load_to_lds)
#define HAVE_TDM 1
#else
#define HAVE_TDM 0
#endif

// ---------------------------------------------------------------------------
// TDM: DMA a 128x128 f32 (64KB) weight matrix from global into LDS.
// D# per CDNA5 ISA ch.8: group0 = {count=1 | lds_addr | global_addr | type=2},
// group1 = {data_size=4B, tensor_dim0/1=128, tile_dim0/1=128, dim0_stride=128}.
// 2-D tensor -> descriptor groups 2/3 unused (zero-filled).
// ---------------------------------------------------------------------------
#if HAVE_TDM
__device__ __forceinline__ void tdm_load_w128(float* lds_dst, const float* gsrc) {
    const unsigned long long ga = (unsigned long long)(uintptr_t)gsrc;
    v4u g0;
    g0[0] = 1u;                                            // count=1, user descriptor
    g0[1] = (unsigned)(uintptr_t)lds_dst;                  // LDS byte address (addr[31:0])
    g0[2] = (unsigned)(ga & 0xffffffffu);                  // global_addr[31:0]
    g0[3] = (unsigned)((ga >> 32) & 0x01ffffffu)           // global_addr[56:32]
          | (2u << 30);                                    // type = 2 ("image")
    v8i g1;
    g1[0] = 0x20000;                 // workgroup_mask=0, data_size=2 (4 bytes)
    g1[1] = (int)(128u << 16);       // tensor_dim0[15:0]=128 into bits[79:64... low half]
    g1[2] = (int)(128u << 16);       // tensor_dim0 hi=0 | tensor_dim1 lo=128
    g1[3] = (int)(128u << 16);       // tensor_dim1 hi=0 | tile_dim0=128
    g1[4] = 128;                     // tile_dim1=128, tile_dim2=0
    g1[5] = 128;                     // tensor_dim0_stride[31:0]=128
    g1[6] = 0;                       // stride hi / dim1_stride lo
    g1[7] = 0;
    const v4i z4 = {};
#if __clang_major__ >= 23
    const v8i z8 = {};
    __builtin_amdgcn_tensor_load_to_lds(g0, g1, z4, z4, z8, 0);
#else
    __builtin_amdgcn_tensor_load_to_lds(g0, g1, z4, z4, 0);
#endif
}
#endif

// Issue the weight stage (wave 0 only) ...
__device__ __forceinline__ void stage_w_issue(float* shW, const float* W, int tid) {
#if HAVE_TDM
    if (tid < 32) tdm_load_w128(shW, W);
#else
    (void)shW; (void)W; (void)tid;
#endif
}
// ... and complete it (TDM wait, or cooperative copy fallback). Caller barriers.
__device__ __forceinline__ void stage_w_finish(float* shW, const float* W, int tid) {
#if HAVE_TDM
    (void)W;
    if (tid < 32) WAIT_TENSORCNT0();
#else
    for (int i = tid; i < (D * D) / 4; i += 256)
        ((float4*)shW)[i] = ((const float4*)W)[i];
#endif
}

// ---------------------------------------------------------------------------
// Edge phase: one wave32 per edge; lane l owns floats [4l,4l+4).
// msg = relu(x[src]+edge_attr); agg[dst] += msg with hardware f32 atomics
// (agg = 25.6MB, L2-resident in the 192MB L2).
// ---------------------------------------------------------------------------
__global__ __launch_bounds__(256) void gine_edge_kernel(
    const float* __restrict__ x, const int* __restrict__ src,
    const int* __restrict__ dst, const float* __restrict__ ea,
    float* __restrict__ agg, int nE)
{
    const int wid  = (int)((blockIdx.x * blockDim.x + threadIdx.x) >> 5); // edge id
    const int lane = threadIdx.x & 31;
    if (wid >= nE) return;                       // wave-uniform exit

    const int s = src[wid];
    const int d = dst[wid];

    // speculative prefetch of a future edge_attr row (global_prefetch_b8)
    if (wid + 64 < nE)
        __builtin_prefetch(ea + (size_t)(wid + 64) * D + lane * 4, 0, 0);

    const float4 xv = *(const float4*)(x  + (size_t)s   * D + lane * 4);
    const float4 ev = *(const float4*)(ea + (size_t)wid * D + lane * 4);

    float4 m;
    m.x = fmaxf(xv.x + ev.x, 0.0f);
    m.y = fmaxf(xv.y + ev.y, 0.0f);
    m.z = fmaxf(xv.z + ev.z, 0.0f);
    m.w = fmaxf(xv.w + ev.w, 0.0f);

    float* b = agg + (size_t)d * D + lane * 4;
    __hip_atomic_fetch_add(b + 0, m.x, __ATOMIC_RELAXED, __HIP_MEMORY_SCOPE_AGENT);
    __hip_atomic_fetch_add(b + 1, m.y, __ATOMIC_RELAXED, __HIP_MEMORY_SCOPE_AGENT);
    __hip_atomic_fetch_add(b + 2, m.z, __ATOMIC_RELAXED, __HIP_MEMORY_SCOPE_AGENT);
    __hip_atomic_fetch_add(b + 3, m.w, __ATOMIC_RELAXED, __HIP_MEMORY_SCOPE_AGENT);
}

// ---------------------------------------------------------------------------
// WMMA tiling: 8 waves/block; wave w owns N-tile w. K loop in steps of 4
// (V_WMMA_F32_16X16X4_F32); B fragment read once per k-step from LDS-staged
// weights, reused across 4 M-tile accumulators (independent WMMAs hide the
// D->A/B RAW hazard).
// A 16x4 f32 (ISA 7.12.2): lanes 0-15 -> M=lane,K={0,1}; 16-31 -> K={2,3}.
// C/D 16x16 f32: VGPR r, lanes 0-15 -> (M=r,N=lane); lanes 16-31 -> M=r+8.
// ---------------------------------------------------------------------------
#define WMMA_F32(acc, a, b) \
    acc = __builtin_amdgcn_wmma_f32_16x16x4_f32(false, a, false, b, (short)0, acc, false, false)

__global__ __launch_bounds__(256) void gine_mlp_kernel(
    const float* __restrict__ xin, const float* __restrict__ agg,
    const float* __restrict__ W1,  const float* __restrict__ b1,
    const float* __restrict__ W2,  const float* __restrict__ b2,
    const float* __restrict__ gamma, const float* __restrict__ beta,
    float* __restrict__ xout, int n)
{
    __shared__ float shA[MB * LSTR];   // x+agg tile; reused for GEMM2 output
    __shared__ float shT[MB * LSTR];   // relu(GEMM1) intermediate
    __shared__ float shW[D * D];       // TDM-staged weight matrix (64KB)

    const int tid  = threadIdx.x;
    const int row0 = blockIdx.x * MB;

    // kick off W1 DMA into LDS; it runs while we stage the activation tile
    stage_w_issue(shW, W1, tid);

    // ---- stage (x + agg) tile into LDS, zero-padding rows past n ----------
    for (int i = tid; i < MB * (D / 4); i += 256) {
        const int r  = i >> 5;          // 32 float4 per row
        const int c4 = i & 31;
        const int g  = row0 + r;
        float4 v = make_float4(0.f, 0.f, 0.f, 0.f);
        if (g < n) {
            const float4 xv = *(const float4*)(xin + (size_t)g * D + c4 * 4);
            const float4 av = *(const float4*)(agg + (size_t)g * D + c4 * 4);
            v.x = xv.x + av.x; v.y = xv.y + av.y;
            v.z = xv.z + av.z; v.w = xv.w + av.w;
        }
        *(float4*)(shA + r * LSTR + c4 * 4) = v;
    }
    stage_w_finish(shW, W1, tid);
    __syncthreads();

    const int wv    = tid >> 5;        // wave id == N-tile id (0..7)
    const int lane  = tid & 31;
    const int lnk   = lane & 15;
    const int khalf = lane >> 4;       // 0: K+{0,1}, 1: K+{2,3}
    const int ncol  = wv * 16 + lnk;   // output column this lane owns

    // ---- GEMM1: T = relu((x+agg) @ W1 + b1) -------------------------------
    {
        v8f acc0 = {}, acc1 = {}, acc2 = {}, acc3 = {};
        for (int s = 0; s < D / 4; ++s) {
            const int k0 = 4 * s + 2 * khalf;
            v2f bf;
            bf[0] = shW[(k0 + 0) * D + ncol];
            bf[1] = shW[(k0 + 1) * D + ncol];
            const v2f a0 = *(const v2f*)(shA + ( 0 + lnk) * LSTR + k0);
            const v2f a1 = *(const v2f*)(shA + (16 + lnk) * LSTR + k0);
            const v2f a2 = *(const v2f*)(shA + (32 + lnk) * LSTR + k0);
            const v2f a3 = *(const v2f*)(shA + (48 + lnk) * LSTR + k0);
            WMMA_F32(acc0, a0, bf);
            WMMA_F32(acc1, a1, bf);
            WMMA_F32(acc2, a2, bf);
            WMMA_F32(acc3, a3, bf);
        }
        const float bb = b1[ncol];
        #pragma unroll
        for (int r = 0; r < 8; ++r) {
            const int mr = 8 * khalf + r;
            shT[( 0 + mr) * LSTR + ncol] = fmaxf(acc0[r] + bb, 0.0f);
            shT[(16 + mr) * LSTR + ncol] = fmaxf(acc1[r] + bb, 0.0f);
            shT[(32 + mr) * LSTR + ncol] = fmaxf(acc2[r] + bb, 0.0f);
            shT[(48 + mr) * LSTR + ncol] = fmaxf(acc3[r] + bb, 0.0f);
        }
    }
    __syncthreads();      // all waves done reading W1 + shA, shT fully written

    // ---- swap W2 into the same LDS weight buffer --------------------------
    stage_w_issue(shW, W2, tid);
    stage_w_finish(shW, W2, tid);
    __syncthreads();

    // ---- GEMM2: H = T @ W2 + b2 (written back into shA) -------------------
    {
        v8f acc0 = {}, acc1 = {}, acc2 = {}, acc3 = {};
        for (int s = 0; s < D / 4; ++s) {
            const int k0 = 4 * s + 2 * khalf;
            v2f bf;
            bf[0] = shW[(k0 + 0) * D + ncol];
            bf[1] = shW[(k0 + 1) * D + ncol];
            const v2f a0 = *(const v2f*)(shT + ( 0 + lnk) * LSTR + k0);
            const v2f a1 = *(const v2f*)(shT + (16 + lnk) * LSTR + k0);
            const v2f a2 = *(const v2f*)(shT + (32 + lnk) * LSTR + k0);
            const v2f a3 = *(const v2f*)(shT + (48 + lnk) * LSTR + k0);
            WMMA_F32(acc0, a0, bf);
            WMMA_F32(acc1, a1, bf);
            WMMA_F32(acc2, a2, bf);
            WMMA_F32(acc3, a3, bf);
        }
        const float bb = b2[ncol];
        #pragma unroll
        for (int r = 0; r < 8; ++r) {
            const int mr = 8 * khalf + r;
            shA[( 0 + mr) * LSTR + ncol] = acc0[r] + bb;
            shA[(16 + mr) * LSTR + ncol] = acc1[r] + bb;
            shA[(32 + mr) * LSTR + ncol] = acc2[r] + bb;
            shA[(48 + mr) * LSTR + ncol] = acc3[r] + bb;
        }
    }
    __syncthreads();

    // ---- LayerNorm + GELU(erf) + residual ---------------------------------
    // wave w handles rows [8w, 8w+8); lane handles 4 columns.
    const float4 gm = *(const float4*)(gamma + lane * 4);
    const float4 bt = *(const float4*)(beta  + lane * 4);
    for (int i = 0; i < 8; ++i) {
        const int r = wv * 8 + i;
        const int g = row0 + r;
        const float4 h = *(const float4*)(shA + r * LSTR + lane * 4);
        float sum = h.x + h.y + h.z + h.w;
        float sq  = h.x * h.x + h.y * h.y + h.z * h.z + h.w * h.w;
        #pragma unroll
        for (int m = 16; m >= 1; m >>= 1) {
            sum += __shfl_xor(sum, m, 32);
            sq  += __shfl_xor(sq,  m, 32);
        }
        const float mean = sum * (1.0f / (float)D);
        const float var  = sq * (1.0f / (float)D) - mean * mean;
        const float rs   = rsqrtf(var + LN_EPS);
        if (g < n) {
            const float4 xr = *(const float4*)(xin + (size_t)g * D + lane * 4);
            float4 o;
            o.x = gelu_erf((h.x - mean) * rs * gm.x + bt.x) + xr.x;
            o.y = gelu_erf((h.y - mean) * rs * gm.y + bt.y) + xr.y;
            o.z = gelu_erf((h.z - mean) * rs * gm.z + bt.z) + xr.z;
            o.w = gelu_erf((h.w - mean) * rs * gm.w + bt.w) + xr.w;
            *(float4*)(xout + (size_t)g * D + lane * 4) = o;
        }
    }
}

// ---------------------------------------------------------------------------
// Final projection: out = x @ Wout + bout (same WMMA tiling, no epilogue)
// ---------------------------------------------------------------------------
__global__ __launch_bounds__(256) void out_proj_kernel(
    const float* __restrict__ xin, const float* __restrict__ Wout,
    const float* __restrict__ bout, float* __restrict__ out, int n)
{
    __shared__ float shA[MB * LSTR];
    __shared__ float shW[D * D];
    const int tid  = threadIdx.x;
    const int row0 = blockIdx.x * MB;

    stage_w_issue(shW, Wout, tid);
    for (int i = tid; i < MB * (D / 4); i += 256) {
        const int r  = i >> 5;
        const int c4 = i & 31;
        const int g  = row0 + r;
        float4 v = make_float4(0.f, 0.f, 0.f, 0.f);
        if (g < n) v = *(const float4*)(xin + (size_t)g * D + c4 * 4);
        *(float4*)(shA + r * LSTR + c4 * 4) = v;
    }
    stage_w_finish(shW, Wout, tid);
    __syncthreads();

    const int wv    = tid >> 5;
    const int lane  = tid & 31;
    const int lnk   = lane & 15;
    const int khalf = lane >> 4;
    const int ncol  = wv * 16 + lnk;

    v8f acc0 = {}, acc1 = {}, acc2 = {}, acc3 = {};
    for (int s = 0; s < D / 4; ++s) {
        const int k0 = 4 * s + 2 * khalf;
        v2f bf;
        bf[0] = shW[(k0 + 0) * D + ncol];
        bf[1] = shW[(k0 + 1) * D + ncol];
        const v2f a0 = *(const v2f*)(shA + ( 0 + lnk) * LSTR + k0);
        const v2f a1 = *(const v2f*)(shA + (16 + lnk) * LSTR + k0);
        const v2f a2 = *(const v2f*)(shA + (32 + lnk) * LSTR + k0);
        const v2f a3 = *(const v2f*)(shA + (48 + lnk) * LSTR + k0);
        WMMA_F32(acc0, a0, bf);
        WMMA_F32(acc1, a1, bf);
        WMMA_F32(acc2, a2, bf);
        WMMA_F32(acc3, a3, bf);
    }
    const float bb = bout[ncol];
    #pragma unroll
    for (int r = 0; r < 8; ++r) {
        const int mr = 8 * khalf + r;
        int g;
        g = row0 +  0 + mr; if (g < n) out[(size_t)g * D + ncol] = acc0[r] + bb;
        g = row0 + 16 + mr; if (g < n) out[(size_t)g * D + ncol] = acc1[r] + bb;
        g = row0 + 32 + mr; if (g < n) out[(size_t)g * D + ncol] = acc2[r] + bb;
        g = row0 + 48 + mr; if (g < n) out[(size_t)g * D + ncol] = acc3[r] + bb;
    }
}

// ---------------------------------------------------------------------------
extern "C" void kernel_launch(void* const* d_in, const int* in_sizes, int n_in,
                              void* d_out, int out_size, void* d_ws, size_t ws_size,
                              hipStream_t stream) {
    const float* x     = (const float*)d_in[0];   // [N, D]
    const int*   ei    = (const int*)  d_in[1];   // [2, E]
    const float* ea    = (const float*)d_in[2];   // [E, D]
    const float* W1    = (const float*)d_in[3];   // [L, D, D]
    const float* b1    = (const float*)d_in[4];   // [L, D]
    const float* W2    = (const float*)d_in[5];
    const float* b2    = (const float*)d_in[6];
    const float* gamma = (const float*)d_in[7];
    const float* beta  = (const float*)d_in[8];
    const float* Wout  = (const float*)d_in[9];   // [D, D_OUT]
    const float* bout  = (const float*)d_in[10];

    const int n  = in_sizes[0] / D;
    const int nE = in_sizes[1] / 2;
    const int L  = in_sizes[3] / (D * D);

    // workspace: ping-pong node features + aggregation buffer (3 * N * D * 4B)
    float* bufA = (float*)d_ws;
    float* bufB = bufA + (size_t)n * D;
    float* agg  = bufB + (size_t)n * D;

    const int mlpBlocks  = (n + MB - 1) / MB;
    const int edgeBlocks = (nE + 7) / 8;          // 8 waves (edges) per block

    const float* cur = x;
    for (int l = 0; l < L; ++l) {
        hipMemsetAsync(agg, 0, (size_t)n * D * sizeof(float), stream);
        gine_edge_kernel<<<edgeBlocks, 256, 0, stream>>>(cur, ei, ei + nE, ea, agg, nE);
        float* nxt = (l & 1) ? bufB : bufA;
        gine_mlp_kernel<<<mlpBlocks, 256, 0, stream>>>(
            cur, agg,
            W1 + (size_t)l * D * D, b1 + (size_t)l * D,
            W2 + (size_t)l * D * D, b2 + (size_t)l * D,
            gamma + (size_t)l * D, beta + (size_t)l * D,
            nxt, n);
        cur = nxt;
    }
    out_proj_kernel<<<mlpBlocks, 256, 0, stream>>>(cur, Wout, bout, (float*)d_out, n);
}